// AngleFreqEnhance_58239756533847
// MI455X (gfx1250) — compile-verified
//
#include <hip/hip_runtime.h>
#include <hip/hip_bf16.h>
#include <math.h>

typedef __attribute__((ext_vector_type(2))) float v2f;
typedef __attribute__((ext_vector_type(8))) float v8f;

#define PI_F      3.14159265358979323846f
#define BW_F      0.26179938779914946f   // radians(15)
#define ALPHA_F   1.2f
#define N_BINS    180
#define BATCH     8
#define CHN       256
#define MCH       16
#define HH        128
#define WW        128
#define HW        16384                  // 128*128
#define FFT_PITCH 129                    // LDS row pitch (float2) to decouple banks
#define TILE_F2   (HH * FFT_PITCH)       // 16512 float2
#define TILE_BYTES (TILE_F2 * 8)         // 132096 B of LDS per image

// ---------------------------------------------------------------------------
// Kernel 1: x_proj[b,m,p] = sum_c w_in[m,c] * x[b,c,p]      (WMMA f32 16x16x4)
// D tile = 16 m-rows x 16 pixels. A = w_in rows (in-lane K pairs),
// B = x rows (lane-contiguous), D rows stored lane-contiguous.
// ---------------------------------------------------------------------------
__global__ __launch_bounds__(256) void proj_kernel(
    const float* __restrict__ x, const float* __restrict__ w_in,
    float* __restrict__ xp) {
  const int wave = blockIdx.x * 8 + (threadIdx.x >> 5);
  const int lane = threadIdx.x & 31;
  const int half = lane >> 4;            // 0: K=0,1 / rows M=0..7 ; 1: K=2,3 / rows M=8..15
  const int l    = lane & 15;
  const int b    = wave >> 10;           // 1024 pixel-tiles per batch
  const int p0   = (wave & 1023) << 4;

  const float* xb = x + (size_t)b * CHN * HW;
  v8f acc = {};
#pragma unroll 4
  for (int c0 = 0; c0 < CHN; c0 += 4) {
    // A (16x4): lane l holds w_in[l][c0+2h], w_in[l][c0+2h+1]
    const float2 aw = *(const float2*)(w_in + l * CHN + c0 + 2 * half);
    v2f A; A.x = aw.x; A.y = aw.y;
    // B (4x16): lane l holds x[b][c0+2h][p0+l], x[b][c0+2h+1][p0+l]
    const float* bp = xb + (size_t)(c0 + 2 * half) * HW + p0 + l;
    v2f B; B.x = bp[0]; B.y = bp[HW];
    acc = __builtin_amdgcn_wmma_f32_16x16x4_f32(false, A, false, B,
                                                (short)0, acc, false, false);
  }
  // D: vgpr r -> m = r + 8*half, lanes 0..15 -> pixels p0+l
  float* dst = xp + ((size_t)(b * MCH + 8 * half) * HW) + p0 + l;
#pragma unroll
  for (int r = 0; r < 8; ++r) dst[(size_t)r * HW] = acc[r];
}

// ---------------------------------------------------------------------------
// In-LDS 128x128 radix-2 complex FFT (both dims). sign=-1 fwd, +1 inv.
// Row pass: lanes walk butterflies within a row (near-contiguous LDS).
// Col pass: lanes walk x for a fixed column butterfly (contiguous LDS).
// ---------------------------------------------------------------------------
__device__ __forceinline__ void fft2d_lds(float2* a, int tid, float sign) {
  // rows: bit-reverse along x
  for (int idx = tid; idx < HW; idx += 256) {
    const int y = idx >> 7, x = idx & 127;
    const int j = (int)(__brev((unsigned)x) >> 25);
    if (x < j) {
      float2 t = a[y * FFT_PITCH + x];
      a[y * FFT_PITCH + x] = a[y * FFT_PITCH + j];
      a[y * FFT_PITCH + j] = t;
    }
  }
  __syncthreads();
  for (int s = 1; s <= 7; ++s) {
    const int m = 1 << s, half = m >> 1;
    for (int t = tid; t < 8192; t += 256) {
      const int row = t >> 6, j = t & 63;
      const int k = j & (half - 1), grp = j >> (s - 1);
      const int i0 = row * FFT_PITCH + grp * m + k;
      const int i1 = i0 + half;
      float sn, cs;
      __sincosf(sign * PI_F * (float)k / (float)half, &sn, &cs);
      const float2 u = a[i0], v = a[i1];
      const float tr = cs * v.x - sn * v.y;
      const float ti = cs * v.y + sn * v.x;
      a[i0] = make_float2(u.x + tr, u.y + ti);
      a[i1] = make_float2(u.x - tr, u.y - ti);
    }
    __syncthreads();
  }
  // columns: bit-reverse along y
  for (int idx = tid; idx < HW; idx += 256) {
    const int y = idx >> 7, x = idx & 127;
    const int j = (int)(__brev((unsigned)y) >> 25);
    if (y < j) {
      float2 t = a[y * FFT_PITCH + x];
      a[y * FFT_PITCH + x] = a[j * FFT_PITCH + x];
      a[j * FFT_PITCH + x] = t;
    }
  }
  __syncthreads();
  for (int s = 1; s <= 7; ++s) {
    const int m = 1 << s, half = m >> 1;
    for (int t = tid; t < 8192; t += 256) {
      const int j = t >> 7, x = t & 127;
      const int k = j & (half - 1), grp = j >> (s - 1);
      const int y0 = grp * m + k, y1 = y0 + half;
      float sn, cs;
      __sincosf(sign * PI_F * (float)k / (float)half, &sn, &cs);
      const float2 u = a[y0 * FFT_PITCH + x], v = a[y1 * FFT_PITCH + x];
      const float tr = cs * v.x - sn * v.y;
      const float ti = cs * v.y + sn * v.x;
      a[y0 * FFT_PITCH + x] = make_float2(u.x + tr, u.y + ti);
      a[y1 * FFT_PITCH + x] = make_float2(u.x - tr, u.y - ti);
    }
    __syncthreads();
  }
}

// ---------------------------------------------------------------------------
// Kernel 2: forward FFT per (b,m) image + ortho scale + orientation histogram
// ---------------------------------------------------------------------------
__global__ __launch_bounds__(256) void fft_fwd_kernel(
    const float* __restrict__ xp, float2* __restrict__ fftb,
    float* __restrict__ hist) {
  extern __shared__ char smem_raw[];
  float2* a  = (float2*)smem_raw;
  float*  lh = (float*)(smem_raw + TILE_BYTES);
  const int tid = threadIdx.x;
  const int img = blockIdx.x;            // b*16 + m
  const int b   = img >> 4;

  for (int i = tid; i < N_BINS; i += 256) lh[i] = 0.0f;
  const float* src = xp + (size_t)img * HW;
  for (int idx = tid; idx < HW; idx += 256) {
    const int y = idx >> 7, x = idx & 127;
    a[y * FFT_PITCH + x] = make_float2(src[idx], 0.0f);
  }
  __syncthreads();

  fft2d_lds(a, tid, -1.0f);

  const float scale = 1.0f / 128.0f;     // ortho: 1/sqrt(128*128)
  float2* dst = fftb + (size_t)img * HW;
  for (int idx = tid; idx < HW; idx += 256) {
    const int y = idx >> 7, x = idx & 127;
    float2 v = a[y * FFT_PITCH + x];
    v.x *= scale; v.y *= scale;
    dst[idx] = v;
    // fftshift-aware angle/radius of this frequency bin
    const int ys = (y + 64) & 127, xs = (x + 64) & 127;
    const float yy = (float)(ys - 64), xx = (float)(xs - 64);
    if (yy != 0.0f || xx != 0.0f) {
      const float theta = atan2f(yy, xx) + PI_F;
      const float tm = fmodf(theta, PI_F);
      int bi = (int)ceilf(tm * ((float)N_BINS / PI_F)) - 1;   // searchsorted(left)-1
      bi = min(max(bi, 0), N_BINS - 1);
      const float mag = sqrtf(v.x * v.x + v.y * v.y) * (1.0f / (float)MCH);
      atomicAdd(&lh[bi], mag);
    }
  }
  __syncthreads();
  for (int i = tid; i < N_BINS; i += 256) atomicAdd(&hist[b * N_BINS + i], lh[i]);
}

// ---------------------------------------------------------------------------
// Kernel 3: smooth histogram, find peak angle per batch
// ---------------------------------------------------------------------------
__global__ void peak_kernel(const float* __restrict__ hist,
                            float* __restrict__ peak) {
  if (threadIdx.x != 0) return;
  const int b = blockIdx.x;
  const float* h = hist + b * N_BINS;
  float e[N_BINS];
  for (int i = 0; i < N_BINS; ++i) {
    const float lft = (i > 0) ? h[i - 1] : 0.0f;
    const float rgt = (i < N_BINS - 1) ? h[i + 1] : 0.0f;
    e[i] = 0.25f * lft + 0.5f * h[i] + 0.25f * rgt;
  }
  int best = -1; float bv = -INFINITY;
  for (int i = 0; i < N_BINS; ++i) {
    const float le = e[(i + N_BINS - 1) % N_BINS];
    const float ri = e[(i + 1) % N_BINS];
    if (e[i] > le && e[i] > ri && e[i] > bv) { bv = e[i]; best = i; }
  }
  if (best < 0) {
    bv = -INFINITY;
    for (int i = 0; i < N_BINS; ++i) if (e[i] > bv) { bv = e[i]; best = i; }
  }
  peak[b] = ((float)best + 0.5f) * (PI_F / (float)N_BINS);
}

// ---------------------------------------------------------------------------
// Kernel 4: angular-band gain in shifted coords + inverse FFT, write real part
// ---------------------------------------------------------------------------
__global__ __launch_bounds__(256) void gain_ifft_kernel(
    const float2* __restrict__ fftb, const float* __restrict__ peak,
    float* __restrict__ xe) {
  extern __shared__ char smem_raw[];
  float2* a = (float2*)smem_raw;
  const int tid = threadIdx.x;
  const int img = blockIdx.x;
  const int b   = img >> 4;
  const float pa   = peak[b];
  const float perp = fmodf(pa + PI_F * 0.5f, PI_F);

  const float2* src = fftb + (size_t)img * HW;
  for (int idx = tid; idx < HW; idx += 256) {
    const int y = idx >> 7, x = idx & 127;
    float2 v = src[idx];
    const int ys = (y + 64) & 127, xs = (x + 64) & 127;
    const float yy = (float)(ys - 64), xx = (float)(xs - 64);
    float g = 1.0f;
    if (yy != 0.0f || xx != 0.0f) {
      const float theta = atan2f(yy, xx) + PI_F;
      float dm = fabsf(theta - pa);   dm = fminf(dm, PI_F - dm);
      float dp = fabsf(theta - perp); dp = fminf(dp, PI_F - dp);
      if (dm <= BW_F || dp <= BW_F) g = ALPHA_F;
    }
    a[y * FFT_PITCH + x] = make_float2(v.x * g, v.y * g);
  }
  __syncthreads();

  fft2d_lds(a, tid, +1.0f);

  const float scale = 1.0f / 128.0f;     // ortho inverse
  float* dst = xe + (size_t)img * HW;
  for (int idx = tid; idx < HW; idx += 256) {
    const int y = idx >> 7, x = idx & 127;
    dst[idx] = a[y * FFT_PITCH + x].x * scale;
  }
}

// ---------------------------------------------------------------------------
// Kernel 5: out[b,c,p] = x[b,c,p] + sum_m w_out[c,m]*x_enh[b,m,p]  (WMMA)
// D tile = 16 channels x 16 pixels; B (x_enh) fragments reused over 16 chunks.
// ---------------------------------------------------------------------------
__global__ __launch_bounds__(256) void out_kernel(
    const float* __restrict__ x, const float* __restrict__ xe,
    const float* __restrict__ w_out, float* __restrict__ out) {
  const int wave = blockIdx.x * 8 + (threadIdx.x >> 5);
  const int lane = threadIdx.x & 31;
  const int half = lane >> 4;
  const int l    = lane & 15;
  const int b    = wave >> 10;
  const int p0   = (wave & 1023) << 4;

  // Preload B fragments: B[k=m][n=pixel] = x_enh[b][m][p0+n], 4 K-steps
  v2f Bf[4];
  const float* xeb = xe + (size_t)b * MCH * HW + p0 + l;
#pragma unroll
  for (int s = 0; s < 4; ++s) {
    const int k0 = 4 * s + 2 * half;
    Bf[s].x = xeb[(size_t)k0 * HW];
    Bf[s].y = xeb[(size_t)(k0 + 1) * HW];
  }

  const float* xb = x   + (size_t)b * CHN * HW;
  float*       ob = out + (size_t)b * CHN * HW;
  for (int cc = 0; cc < 16; ++cc) {
    const int c0 = cc * 16;
    // C initialized from x -> residual add is free
    v8f acc;
#pragma unroll
    for (int r = 0; r < 8; ++r)
      acc[r] = xb[(size_t)(c0 + r + 8 * half) * HW + p0 + l];
#pragma unroll
    for (int s = 0; s < 4; ++s) {
      const int k0 = 4 * s;
      // A[M=c][K=m] = w_out[c0+l][k0 + 2h (+1)] — contiguous pair per lane
      const float2 aw = *(const float2*)(w_out + (c0 + l) * MCH + k0 + 2 * half);
      v2f A; A.x = aw.x; A.y = aw.y;
      acc = __builtin_amdgcn_wmma_f32_16x16x4_f32(false, A, false, Bf[s],
                                                  (short)0, acc, false, false);
    }
#pragma unroll
    for (int r = 0; r < 8; ++r)
      ob[(size_t)(c0 + r + 8 * half) * HW + p0 + l] = acc[r];
  }
}

__global__ void zero_kernel(float* __restrict__ p, int n) {
  const int i = blockIdx.x * blockDim.x + threadIdx.x;
  if (i < n) p[i] = 0.0f;
}

// ---------------------------------------------------------------------------
extern "C" void kernel_launch(void* const* d_in, const int* in_sizes, int n_in,
                              void* d_out, int out_size, void* d_ws, size_t ws_size,
                              hipStream_t stream) {
  const float* x     = (const float*)d_in[0];   // (8,256,128,128)
  const float* w_in  = (const float*)d_in[1];   // (16,256)
  const float* w_out = (const float*)d_in[2];   // (256,16)
  float*       out   = (float*)d_out;

  char* ws = (char*)d_ws;
  const size_t XPROJ_BYTES = (size_t)BATCH * MCH * HW * sizeof(float);   // 8 MB
  const size_t FFT_BYTES   = (size_t)BATCH * MCH * HW * sizeof(float2);  // 16 MB
  const size_t XENH_BYTES  = XPROJ_BYTES;                                // 8 MB
  float*  x_proj = (float*)ws;
  float2* fftb   = (float2*)(ws + XPROJ_BYTES);
  float*  x_enh  = (float*)(ws + XPROJ_BYTES + FFT_BYTES);
  float*  hist   = (float*)(ws + XPROJ_BYTES + FFT_BYTES + XENH_BYTES);
  float*  peak   = hist + BATCH * N_BINS;

  const int n_hist = BATCH * N_BINS;
  zero_kernel<<<(n_hist + 255) / 256, 256, 0, stream>>>(hist, n_hist);

  // 8192 waves total (16 pixels each), 8 waves/block
  proj_kernel<<<1024, 256, 0, stream>>>(x, w_in, x_proj);

  const size_t fwd_lds = TILE_BYTES + N_BINS * sizeof(float);
  fft_fwd_kernel<<<BATCH * MCH, 256, fwd_lds, stream>>>(x_proj, fftb, hist);

  peak_kernel<<<BATCH, 32, 0, stream>>>(hist, peak);

  gain_ifft_kernel<<<BATCH * MCH, 256, (size_t)TILE_BYTES, stream>>>(fftb, peak, x_enh);

  out_kernel<<<1024, 256, 0, stream>>>(x, x_enh, w_out, out);
}